// GNNModel_65627100283423
// MI455X (gfx1250) — compile-verified
//
#include <hip/hip_runtime.h>
#include <hip/hip_bf16.h>

// ---------------------------------------------------------------------------
// Heterogeneous GINE GNN forward for MI455X (gfx1250, wave32, WMMA).
// Dense GEMMs use V_WMMA_F32_16X16X4_F32 (fp32 matrix path, matches the fp32
// reference bit-for-bit semantics up to summation order). K is templated
// (3 / 13 / 128) so the WMMA K-loop is completely branch-free; A tiles are
// staged in LDS with coalesced float4 loads and a 132-float row stride for
// conflict-free ds_load_b64 fragment reads. Edge linear (K=4) is fused into
// the gather/scatter kernel; segment-sum uses f32 global atomics that execute
// in the 192MB L2 (agg arrays are 51MB).
// ---------------------------------------------------------------------------

typedef float v2f __attribute__((ext_vector_type(2)));
typedef float v8f __attribute__((ext_vector_type(8)));
typedef float f4v __attribute__((ext_vector_type(4)));

#define NCC 100000
#define NAA 20000
#define NEE 600000
#define HH  128

__device__ __forceinline__ v8f wmma_f32_4(v2f a, v2f b, v8f c) {
    return __builtin_amdgcn_wmma_f32_16x16x4_f32(
        /*neg_a=*/false, a, /*neg_b=*/false, b,
        /*c_mod=*/(short)0, c, /*reuse_a=*/false, /*reuse_b=*/false);
}

// ---------------------------------------------------------------------------
// K=128 GEMM: C[M,128] = act(A[M,128] @ W[128,128] + bias).
// Block = 256 threads = 8 waves; block owns a 16-row tile, wave w owns
// columns [16w,16w+16). A tile staged in LDS (16x132 padded, coalesced
// float4 loads, row clamped to M-1 so no exec branching); B read direct
// (64KB weight matrix is L2-resident). Inner loop: 32 unpredicated WMMAs.
// ---------------------------------------------------------------------------
template<int ACT>
__global__ __launch_bounds__(256)
void gemm_k128(const float* __restrict__ A, const float* __restrict__ W,
               const float* __restrict__ bias, float* __restrict__ C, int M)
{
    __shared__ float sA[16 * 132];

    const int tile_m = blockIdx.x * 16;
    const int t = threadIdx.x;

    // Cooperative A-tile load: 2048 floats as 512 float4s, 2 per thread.
#pragma unroll
    for (int i = 0; i < 2; ++i) {
        const int idx = (t + i * 256) * 4;       // flat float index in tile
        const int row = idx >> 7;                // /128
        const int col = idx & 127;
        const int gr  = min(tile_m + row, M - 1);
        const f4v v = *(const f4v*)(A + (size_t)gr * HH + col);
        *(f4v*)(&sA[row * 132 + col]) = v;
    }
    __syncthreads();

    const int lane = t & 31;
    const int wave = t >> 5;
    const int hi = lane >> 4;
    const int lo = lane & 15;
    const int n  = wave * 16 + lo;               // output column
    const float* __restrict__ wp = W + n;

    v8f c = {0.f, 0.f, 0.f, 0.f, 0.f, 0.f, 0.f, 0.f};
#pragma unroll
    for (int k = 0; k < HH; k += 4) {
        const int kc = k + 2 * hi;
        const v2f a = *(const v2f*)(&sA[lo * 132 + kc]);   // ds_load_b64
        v2f b;
        b[0] = wp[(size_t)kc * HH];
        b[1] = wp[(size_t)(kc + 1) * HH];
        c = wmma_f32_4(a, b, c);
    }

    const float bv = bias[n];
#pragma unroll
    for (int r = 0; r < 8; ++r) {
        const int mr = tile_m + r + 8 * hi;
        if (mr < M) {
            float v = c[r] + bv;
            if (ACT) v = fmaxf(v, 0.f);
            C[(size_t)mr * HH + n] = v;
        }
    }
}

// ---------------------------------------------------------------------------
// Small-K GEMM (K = 3 or 13): both A (16 x Kp, zero-padded, stride Kp+2) and
// W (Kp x 128, zero-padded) staged in LDS; inner product is 1-4 branch-free
// WMMAs. Used for the input feature linears.
// ---------------------------------------------------------------------------
template<int K, int ACT>
__global__ __launch_bounds__(256)
void gemm_smallk(const float* __restrict__ A, const float* __restrict__ W,
                 const float* __restrict__ bias, float* __restrict__ C, int M)
{
    constexpr int Kp = (K + 3) & ~3;
    constexpr int AS = Kp + 2;                   // padded LDS row stride
    __shared__ float sA[16 * AS];
    __shared__ float sW[Kp * HH];

    const int tile_m = blockIdx.x * 16;
    const int t = threadIdx.x;

    // Stage W, zero-padding rows K..Kp-1.
    for (int idx = t; idx < Kp * HH; idx += 256) {
        const int row = idx >> 7;
        const int col = idx & 127;
        sW[idx] = (row < K) ? W[(size_t)row * HH + col] : 0.f;
    }
    // Stage A tile, zero-padding cols K..Kp-1, clamped rows.
    if (t < 16 * Kp) {
        const int row = t / Kp;
        const int col = t - row * Kp;
        const int gr  = min(tile_m + row, M - 1);
        sA[row * AS + col] = (col < K) ? A[(size_t)gr * K + col] : 0.f;
    }
    __syncthreads();

    const int lane = t & 31;
    const int wave = t >> 5;
    const int hi = lane >> 4;
    const int lo = lane & 15;
    const int n  = wave * 16 + lo;

    v8f c = {0.f, 0.f, 0.f, 0.f, 0.f, 0.f, 0.f, 0.f};
#pragma unroll
    for (int k = 0; k < Kp; k += 4) {
        const int kc = k + 2 * hi;
        v2f a, b;
        a[0] = sA[lo * AS + kc];
        a[1] = sA[lo * AS + kc + 1];
        b[0] = sW[kc * HH + n];
        b[1] = sW[(kc + 1) * HH + n];
        c = wmma_f32_4(a, b, c);
    }

    const float bv = bias[n];
#pragma unroll
    for (int r = 0; r < 8; ++r) {
        const int mr = tile_m + r + 8 * hi;
        if (mr < M) {
            float v = c[r] + bv;
            if (ACT) v = fmaxf(v, 0.f);
            C[(size_t)mr * HH + n] = v;
        }
    }
}

// ---------------------------------------------------------------------------
// Fused edge kernel: for each edge e,
//   msg[f] = relu( xsrc[f] + eb[f] + sum_k edge_attr[e][k]*ew[k][f] )
//   agg[dst[e]][f] += msg[f]   (f32 global atomic, executes in L2)
// BCAST selects the layer-2 broadcast mode ([NA,1] source).
// One wave per edge; lane covers 4 features (float4). ew/eb staged in LDS.
// ---------------------------------------------------------------------------
template<int BCAST>
__global__ __launch_bounds__(256)
void edge_scatter(const float* __restrict__ xs,
                  const float* __restrict__ edge_attr,
                  const int* __restrict__ sidx, const int* __restrict__ didx,
                  const float* __restrict__ ew, const float* __restrict__ eb,
                  float* __restrict__ agg, int nE)
{
    __shared__ float s_ew[4 * HH];
    __shared__ float s_eb[HH];
    for (int i = threadIdx.x; i < 4 * HH; i += 256) s_ew[i] = ew[i];
    if (threadIdx.x < HH) s_eb[threadIdx.x] = eb[threadIdx.x];
    __syncthreads();

    const int lane = threadIdx.x & 31;
    const int wave = threadIdx.x >> 5;
    const int f0 = lane * 4;
    const int stride = gridDim.x * 8;

    for (int e = blockIdx.x * 8 + wave; e < nE; e += stride) {
        const int s = sidx[e];
        const int d = didx[e];
        const f4v ea = *(const f4v*)(edge_attr + (size_t)e * 4);
        f4v xv;
        if (BCAST) {
            const float v = xs[s];
            xv = (f4v){v, v, v, v};
        } else {
            xv = *(const f4v*)(xs + (size_t)s * HH + f0);
        }
        float* ag = agg + (size_t)d * HH + f0;
#pragma unroll
        for (int j = 0; j < 4; ++j) {
            const int f = f0 + j;
            float msg = xv[j] + s_eb[f]
                      + ea.x * s_ew[f]
                      + ea.y * s_ew[HH + f]
                      + ea.z * s_ew[2 * HH + f]
                      + ea.w * s_ew[3 * HH + f];
            msg = fmaxf(msg, 0.f);
            __hip_atomic_fetch_add(ag + j, msg, __ATOMIC_RELAXED,
                                   __HIP_MEMORY_SCOPE_AGENT);
        }
    }
}

// ---------------------------------------------------------------------------
// Elementwise helpers
// ---------------------------------------------------------------------------
__global__ __launch_bounds__(256)
void zero_f32(float* __restrict__ p, size_t n)
{
    for (size_t i = blockIdx.x * 256ull + threadIdx.x; i < n; i += gridDim.x * 256ull)
        p[i] = 0.f;
}

// h = (1+eps)*xd + agg      (eps read on-device from 1-element array)
__global__ __launch_bounds__(256)
void combine_eps(const float* __restrict__ xd, const float* __restrict__ agg,
                 const float* __restrict__ eps, float* __restrict__ h, size_t n)
{
    const float s = 1.0f + eps[0];
    for (size_t i = blockIdx.x * 256ull + threadIdx.x; i < n; i += gridDim.x * 256ull)
        h[i] = s * xd[i] + agg[i];
}

// x = relu(x * gamma/sqrt(1+bn_eps) + beta)   (eval-mode BN, mean 0 / var 1)
__global__ __launch_bounds__(256)
void bn_relu(float* __restrict__ x, const float* __restrict__ gamma,
             const float* __restrict__ beta, size_t n)
{
    const float inv = 0.99999500003749968f;   // 1/sqrt(1 + 1e-5)
    for (size_t i = blockIdx.x * 256ull + threadIdx.x; i < n; i += gridDim.x * 256ull) {
        const int f = (int)(i & (HH - 1));
        x[i] = fmaxf(x[i] * gamma[f] * inv + beta[f], 0.f);
    }
}

// out_sig[row] = sigmoid(dot(X[row,:128], w2) + b2); optional raw copy.
// One wave per row; 4 elems/lane; wave32 shfl_xor reduction.
__global__ __launch_bounds__(256)
void rowdot_sigmoid(const float* __restrict__ X, const float* __restrict__ w2,
                    const float* __restrict__ b2, float* __restrict__ out_sig,
                    float* __restrict__ out_raw, int M)
{
    const int lane = threadIdx.x & 31;
    const int wave = threadIdx.x >> 5;
    const int row = blockIdx.x * 8 + wave;
    if (row >= M) return;
    const float* x = X + (size_t)row * HH;
    float sum = 0.f;
#pragma unroll
    for (int j = 0; j < 4; ++j)
        sum += x[lane + j * 32] * w2[lane + j * 32];
    for (int off = 16; off > 0; off >>= 1)
        sum += __shfl_xor(sum, off, 32);
    if (lane == 0) {
        const float v = sum + b2[0];
        if (out_raw) out_raw[row] = v;
        out_sig[row] = 1.f / (1.f + __expf(-v));
    }
}

// ---------------------------------------------------------------------------
// Host-side orchestration
// ---------------------------------------------------------------------------
static inline int cdiv(int a, int b) { return (a + b - 1) / b; }

extern "C" void kernel_launch(void* const* d_in, const int* in_sizes, int n_in,
                              void* d_out, int out_size, void* d_ws, size_t ws_size,
                              hipStream_t stream)
{
    // ---- input pytree leaves (insertion order, params flattened key-sorted) ----
    const float* x_company = (const float*)d_in[0];   // [NC,13]
    const float* x_analyst = (const float*)d_in[1];   // [NA,3]
    const int*   edge_src  = (const int*)  d_in[2];   // [NE]
    const int*   edge_dst  = (const int*)  d_in[3];   // [NE]
    const float* edge_attr = (const float*)d_in[4];   // [NE,4]
    const float* bn_beta[2]  = { (const float*)d_in[5], (const float*)d_in[7] };
    const float* bn_gamma[2] = { (const float*)d_in[6], (const float*)d_in[8] };
    // layer 0 leaves: ab aw b1 b2 cb cw eb eps ew w1 w2  -> indices 9..19
    // layer 1 leaves: same                              -> indices 20..30
    // layer 2 leaves: b1 b2 eb eps ew w1 w2             -> indices 31..37
    const float *ab[2], *aw[2], *cb[2], *cw[2];
    const float *b1[3], *b2[3], *eb[3], *epsp[3], *ew[3], *w1[3], *w2[3];
    for (int l = 0; l < 2; ++l) {
        const int base = 9 + l * 11;
        ab[l]   = (const float*)d_in[base + 0];
        aw[l]   = (const float*)d_in[base + 1];
        b1[l]   = (const float*)d_in[base + 2];
        b2[l]   = (const float*)d_in[base + 3];
        cb[l]   = (const float*)d_in[base + 4];
        cw[l]   = (const float*)d_in[base + 5];
        eb[l]   = (const float*)d_in[base + 6];
        epsp[l] = (const float*)d_in[base + 7];
        ew[l]   = (const float*)d_in[base + 8];
        w1[l]   = (const float*)d_in[base + 9];
        w2[l]   = (const float*)d_in[base + 10];
    }
    b1[2]   = (const float*)d_in[31];
    b2[2]   = (const float*)d_in[32];
    eb[2]   = (const float*)d_in[33];
    epsp[2] = (const float*)d_in[34];
    ew[2]   = (const float*)d_in[35];
    w1[2]   = (const float*)d_in[36];
    w2[2]   = (const float*)d_in[37];

    float* dout = (float*)d_out;   // [NC + NA], xc sigmoid first, then xa

    // ---- workspace carve (3x[NC,128] + 3x[NA,128] fp32 ~= 184 MB) ----
    const size_t NCH = (size_t)NCC * HH;
    const size_t NAH = (size_t)NAA * HH;
    float* ws = (float*)d_ws;
    float* c0 = ws;            // company buffers
    float* c1 = c0 + NCH;
    float* c2 = c1 + NCH;
    float* a0 = c2 + NCH;      // analyst buffers
    float* a1 = a0 + NAH;
    float* a2 = a1 + NAH;
    (void)ws_size; (void)n_in; (void)in_sizes; (void)out_size;

    const dim3 blk(256);
    const int EW_BLOCKS = 2048;   // grid-stride over 600k edges, 8 edges/block/iter
    const int EL_BLOCKS = 4096;   // elementwise grid-stride

    auto gemm128 = [&](const float* A, const float* W, const float* B, float* C,
                       int M, int act) {
        if (act)
            gemm_k128<1><<<dim3(cdiv(M, 16)), blk, 0, stream>>>(A, W, B, C, M);
        else
            gemm_k128<0><<<dim3(cdiv(M, 16)), blk, 0, stream>>>(A, W, B, C, M);
    };
    auto scat = [&](const float* xs, const int* s, const int* d,
                    const float* w, const float* b, float* agg) {
        edge_scatter<0><<<EW_BLOCKS, blk, 0, stream>>>(xs, edge_attr, s, d, w, b,
                                                       agg, NEE);
    };

    // ========================== layer 0 ==========================
    gemm_smallk<13, 0><<<dim3(cdiv(NCC, 16)), blk, 0, stream>>>(
        x_company, cw[0], cb[0], c1, NCC);                               // xc lin -> c1
    gemm_smallk<3, 0><<<dim3(cdiv(NAA, 16)), blk, 0, stream>>>(
        x_analyst, aw[0], ab[0], a1, NAA);                               // xa lin -> a1
    zero_f32<<<EL_BLOCKS, blk, 0, stream>>>(a2, NAH);
    scat(c1, edge_src, edge_dst, ew[0], eb[0], a2);                      // agg_a -> a2
    combine_eps<<<EL_BLOCKS, blk, 0, stream>>>(a1, a2, epsp[0], a0, NAH);// h_a -> a0
    gemm128(a0, w1[0], b1[0], a2, NAA, 1);                               // relu mlp1 -> a2
    gemm128(a2, w2[0], b2[0], a1, NAA, 0);                               // new xa -> a1
    zero_f32<<<EL_BLOCKS, blk, 0, stream>>>(c2, NCH);
    scat(a1, edge_dst, edge_src, ew[0], eb[0], c2);                      // agg_c -> c2
    combine_eps<<<EL_BLOCKS, blk, 0, stream>>>(c1, c2, epsp[0], c0, NCH);// h_c -> c0
    gemm128(c0, w1[0], b1[0], c2, NCC, 1);
    gemm128(c2, w2[0], b2[0], c1, NCC, 0);                               // new xc -> c1
    bn_relu<<<EL_BLOCKS, blk, 0, stream>>>(c1, bn_gamma[0], bn_beta[0], NCH);
    bn_relu<<<EL_BLOCKS, blk, 0, stream>>>(a1, bn_gamma[0], bn_beta[0], NAH);

    // ========================== layer 1 ==========================
    gemm128(c1, cw[1], cb[1], c0, NCC, 0);                               // xc lin -> c0
    gemm128(a1, aw[1], ab[1], a0, NAA, 0);                               // xa lin -> a0
    zero_f32<<<EL_BLOCKS, blk, 0, stream>>>(a2, NAH);
    scat(c0, edge_src, edge_dst, ew[1], eb[1], a2);
    combine_eps<<<EL_BLOCKS, blk, 0, stream>>>(a0, a2, epsp[1], a1, NAH);// h_a -> a1
    gemm128(a1, w1[1], b1[1], a2, NAA, 1);
    gemm128(a2, w2[1], b2[1], a0, NAA, 0);                               // new xa -> a0
    zero_f32<<<EL_BLOCKS, blk, 0, stream>>>(c2, NCH);
    scat(a0, edge_dst, edge_src, ew[1], eb[1], c2);
    combine_eps<<<EL_BLOCKS, blk, 0, stream>>>(c0, c2, epsp[1], c1, NCH);// h_c -> c1
    gemm128(c1, w1[1], b1[1], c2, NCC, 1);
    gemm128(c2, w2[1], b2[1], c0, NCC, 0);                               // new xc -> c0
    bn_relu<<<EL_BLOCKS, blk, 0, stream>>>(c0, bn_gamma[1], bn_beta[1], NCH);
    bn_relu<<<EL_BLOCKS, blk, 0, stream>>>(a0, bn_gamma[1], bn_beta[1], NAH);

    // ========================== layer 2 ==========================
    // xc in c0, xa in a0
    zero_f32<<<EL_BLOCKS, blk, 0, stream>>>(a2, NAH);
    scat(c0, edge_src, edge_dst, ew[2], eb[2], a2);
    combine_eps<<<EL_BLOCKS, blk, 0, stream>>>(a0, a2, epsp[2], a1, NAH);// h_a -> a1
    gemm128(a1, w1[2], b1[2], a2, NAA, 1);                               // relu mlp1 -> a2
    // xa2 = a2 @ w2 + b2 : raw to a1[0..NA), sigmoid to dout[NC..NC+NA)
    rowdot_sigmoid<<<dim3(cdiv(NAA, 8)), blk, 0, stream>>>(a2, w2[2], b2[2],
                  dout + NCC, a1, NAA);
    zero_f32<<<EL_BLOCKS, blk, 0, stream>>>(c2, NCH);
    // reverse conv: source is scalar xa2 broadcast over 128 features
    edge_scatter<1><<<EW_BLOCKS, blk, 0, stream>>>(a1, edge_attr,
                  edge_dst, edge_src, ew[2], eb[2], c2, NEE);
    combine_eps<<<EL_BLOCKS, blk, 0, stream>>>(c0, c2, epsp[2], c1, NCH);// h_c -> c1
    gemm128(c1, w1[2], b1[2], c2, NCC, 1);
    rowdot_sigmoid<<<dim3(cdiv(NCC, 8)), blk, 0, stream>>>(c2, w2[2], b2[2],
                  dout, nullptr, NCC);
}